// SequenceAttLayer_73031623901807
// MI455X (gfx1250) — compile-verified
//
#include <hip/hip_runtime.h>
#include <math.h>

typedef __attribute__((ext_vector_type(2))) float v2f;
typedef __attribute__((ext_vector_type(8))) float v8f;

#define B_    1024
#define S_    256
#define D_    128
#define H1_   256
#define H2_   64
#define CH_   32            // sequence rows per chunk
#define NCH   (S_ / CH_)    // 8 chunks
#define SKP   132           // padded LDS stride for sK (132 % 64 = 4 -> conflict-free)
#define SHP   260           // padded LDS stride for sH

__device__ __forceinline__ v8f wmma_f32(v2f a, v2f b, v8f c) {
  // D = A(16x4) * B(4x16) + C, exact fp32 on the CDNA5 matrix pipe
  return __builtin_amdgcn_wmma_f32_16x16x4_f32(false, a, false, b, (short)0, c,
                                               false, false);
}

// Repack weights into WMMA-B-fragment-ready pairs (and fold the concat algebra):
//  Pk [p*256+n] = (Ak[2p][n],  Ak[2p+1][n]),  Ak  = W1[128+d] - W1[256+d]
//  Pqk[p*256+n] = (Aqk[2p][n], Aqk[2p+1][n]), Aqk = W1[384+d]
//  P2 [p*64 +n] = (W2[2p][n],  W2[2p+1][n])
//  Aq [d*256+n] = W1[d][n] + W1[256+d][n]        (for the per-batch Q1 vector)
__global__ void din_prep(const float* __restrict__ W1,
                         const float* __restrict__ W2,
                         float2* __restrict__ Pk,
                         float2* __restrict__ Pqk,
                         float2* __restrict__ P2,
                         float*  __restrict__ Aq) {
  int idx = blockIdx.x * blockDim.x + threadIdx.x;
  if (idx < 64 * 256) {
    int p = idx >> 8, n = idx & 255;
    int k = 2 * p;
    float a0 = W1[(128 + k) * H1_ + n]     - W1[(256 + k) * H1_ + n];
    float a1 = W1[(128 + k + 1) * H1_ + n] - W1[(256 + k + 1) * H1_ + n];
    Pk[idx]  = make_float2(a0, a1);
    Pqk[idx] = make_float2(W1[(384 + k) * H1_ + n], W1[(384 + k + 1) * H1_ + n]);
  } else if (idx < 64 * 256 + 128 * 64) {
    int j = idx - 64 * 256;
    int p = j >> 6, n = j & 63;
    int k = 2 * p;
    P2[j] = make_float2(W2[k * H2_ + n], W2[(k + 1) * H2_ + n]);
  } else {
    int j = idx - (64 * 256 + 128 * 64);
    if (j < 128 * 256)                      // flat j = d*256 + n
      Aq[j] = W1[j] + W1[256 * 256 + j];
  }
}

__global__ __launch_bounds__(256)
void din_attn(const float* __restrict__ q_g,
              const float* __restrict__ keys,
              const int*   __restrict__ klen,
              const float* __restrict__ Aq,
              const float* __restrict__ b1,
              const float* __restrict__ b2,
              const float* __restrict__ Wd,
              const float* __restrict__ bd,
              const float2* __restrict__ Pk,
              const float2* __restrict__ Pqk,
              const float2* __restrict__ P2,
              float* __restrict__ out) {
  __shared__ float sQ[D_];
  __shared__ float sQ1[H1_];
  __shared__ float sScore[S_];
  __shared__ float sRed;
  __shared__ alignas(16) float sK[CH_ * SKP];
  __shared__ alignas(16) float sH[CH_ * SHP];

  const int b    = blockIdx.x;
  const int tid  = threadIdx.x;
  const int lane = tid & 31;
  const int wv   = tid >> 5;
  const int ln15 = lane & 15;
  const int hi   = (lane & 16) ? 1 : 0;   // upper half-wave -> K offset +2, M offset +8

  sScore[tid] = bd[0];
  if (tid < D_) sQ[tid] = q_g[b * D_ + tid];
  __syncthreads();

  // Q1[h] = b1[h] + sum_d q[d]*Aq[d,h]   (folded q and q-k concat terms)
  {
    float s = b1[tid];
    for (int d = 0; d < D_; ++d)
      s += sQ[d] * Aq[d * H1_ + tid];
    sQ1[tid] = s;
  }
  __syncthreads();

  const float* kb = keys + (size_t)b * S_ * D_;

  for (int c = 0; c < NCH; ++c) {
    // ---- stage 32x128 keys chunk into LDS (padded rows) ----
    {
      int r = tid >> 3, q8 = tid & 7;               // 8 threads per row
      const float4* src = (const float4*)(kb + c * CH_ * D_ + r * D_);
      float4* dst = (float4*)(sK + r * SKP);
      for (int i = 0; i < 4; ++i) dst[q8 + i * 8] = src[q8 + i * 8];
    }
    if (c + 1 < NCH)
      __builtin_prefetch(kb + (c + 1) * CH_ * D_ + tid * 16, 0, 0);
    __syncthreads();

    // ---- GEMM1: H1pre[32x256] = K@Ak + (q.K)@Aqk, K-dim 128 (f32 WMMA) ----
    // Each wave: 2 row-tiles x 2 col-tiles -> B fragments amortize over 32 rows.
    {
      const int ct0  = wv * 2;             // this wave's two col tiles
      const int koff = hi * 2;
      const int ar0  = ln15;               // row-tile 0 row
      const int ar1  = 16 + ln15;          // row-tile 1 row
      v8f z = {0,0,0,0,0,0,0,0};
      v8f a00 = z, a01 = z, a10 = z, a11 = z;   // acc[rt][cti]
      for (int k0 = 0; k0 < D_; k0 += 4) {
        v2f aK0 = *(const v2f*)&sK[ar0 * SKP + k0 + koff];
        v2f aK1 = *(const v2f*)&sK[ar1 * SKP + k0 + koff];
        v2f qp  = *(const v2f*)&sQ[k0 + koff];
        v2f aQ0 = aK0 * qp;
        v2f aQ1 = aK1 * qp;
        const int prow = (k0 >> 1) + hi;
        const v2f* bk = (const v2f*)(Pk  + prow * H1_);
        const v2f* bq = (const v2f*)(Pqk + prow * H1_);
        const int n0 = ct0 * 16 + ln15;
        v2f b0 = bk[n0], b1v = bk[n0 + 16];
        v2f c0 = bq[n0], c1v = bq[n0 + 16];
        a00 = wmma_f32(aK0, b0,  a00);  a10 = wmma_f32(aK1, b0,  a10);
        a01 = wmma_f32(aK0, b1v, a01);  a11 = wmma_f32(aK1, b1v, a11);
        a00 = wmma_f32(aQ0, c0,  a00);  a10 = wmma_f32(aQ1, c0,  a10);
        a01 = wmma_f32(aQ0, c1v, a01);  a11 = wmma_f32(aQ1, c1v, a11);
      }
      // bias + sigmoid, write to sH
      v8f accs[2][2] = {{a00, a01}, {a10, a11}};
      for (int rt = 0; rt < 2; ++rt) {
        for (int i = 0; i < 2; ++i) {
          int col = (ct0 + i) * 16 + ln15;
          float q1 = sQ1[col];
          for (int j = 0; j < 8; ++j) {
            int row = rt * 16 + j + hi * 8;
            float x = accs[rt][i][j] + q1;
            sH[row * SHP + col] = 1.0f / (1.0f + __expf(-x));   // sigmoid
          }
        }
      }
    }
    __syncthreads();

    // ---- GEMM2: H2[32x64] = sH @ W2 over K-dim 256; score += sigmoid(.)·Wd ----
    {
      const int rt2  = wv >> 2;
      const int ct2  = wv & 3;
      const int arow = rt2 * 16 + ln15;
      const int koff = hi * 2;
      const int n2   = ct2 * 16 + ln15;
      v8f acc = {0,0,0,0,0,0,0,0};
      for (int k0 = 0; k0 < H1_; k0 += 4) {
        v2f a  = *(const v2f*)&sH[arow * SHP + k0 + koff];
        v2f bv = ((const v2f*)P2)[((k0 >> 1) + hi) * H2_ + n2];
        acc = wmma_f32(a, bv, acc);
      }
      float bias = b2[n2];
      float wd   = Wd[n2];
      for (int j = 0; j < 8; ++j) {
        int row = rt2 * 16 + j + hi * 8;
        float h2 = 1.0f / (1.0f + __expf(-(acc[j] + bias)));
        atomicAdd(&sScore[c * CH_ + row], h2 * wd);   // ds_add_f32
      }
    }
    __syncthreads();
  }

  // ---- mask, scale by 1/sqrt(D), softmax ----
  {
    int kl = klen[b];
    float raw = sScore[tid];
    sScore[tid] = (tid < kl) ? raw * 0.08838834764831845f : -INFINITY;
  }
  __syncthreads();
  if (tid == 0) {
    float m = -INFINITY;
    for (int s = 0; s < S_; ++s) m = fmaxf(m, sScore[s]);
    float l = 0.0f;
    for (int s = 0; s < S_; ++s) {
      float p = __expf(sScore[s] - m);
      sScore[s] = p;
      l += p;
    }
    sRed = 1.0f / l;
  }
  __syncthreads();

  // ---- weighted sum of keys: out[b,d] = sum_s w_s * keys[b,s,d] ----
  if (tid < D_) {
    const float* kp = kb + tid;
    float acc = 0.0f;
    for (int s = 0; s < S_; ++s) acc += sScore[s] * kp[s * D_];
    out[b * D_ + tid] = acc * sRed;
  }
}

extern "C" void kernel_launch(void* const* d_in, const int* in_sizes, int n_in,
                              void* d_out, int out_size, void* d_ws, size_t ws_size,
                              hipStream_t stream) {
  const float* q    = (const float*)d_in[0];
  const float* keys = (const float*)d_in[1];
  const int*   kl   = (const int*)d_in[2];
  const float* W1   = (const float*)d_in[3];
  const float* b1   = (const float*)d_in[4];
  const float* W2   = (const float*)d_in[5];
  const float* b2   = (const float*)d_in[6];
  const float* Wd   = (const float*)d_in[7];
  const float* bd   = (const float*)d_in[8];

  float2* Pk  = (float2*)d_ws;           // 64*256 float2 = 128 KB
  float2* Pqk = Pk  + 64 * 256;          // 128 KB
  float2* P2  = Pqk + 64 * 256;          // 128*64 float2 = 64 KB
  float*  Aq  = (float*)(P2 + 128 * 64); // 128*256 f32  = 128 KB  (total 448 KB)

  // 16384 (Pk/Pqk pairs) + 8192 (P2) + 32768 (Aq) = 57344 threads -> 224 blocks
  din_prep<<<224, 256, 0, stream>>>(W1, W2, Pk, Pqk, P2, Aq);
  din_attn<<<B_, 256, 0, stream>>>(q, keys, kl, Aq, b1, b2, Wd, bd,
                                   Pk, Pqk, P2, (float*)d_out);
}